// QR_49263274885245
// MI455X (gfx1250) — compile-verified
//
#include <hip/hip_runtime.h>

typedef float v2f __attribute__((ext_vector_type(2)));
typedef float v8f __attribute__((ext_vector_type(8)));

#define ZDIM 128

// x + row_ror:N(x)  — DPP16 rotation within each 16-lane row (pure VALU, no LDS)
template<int N>
__device__ __forceinline__ float rotadd16(float x) {
    int yi = __builtin_amdgcn_update_dpp(0, __float_as_int(x),
                                         0x120 | N, 0xF, 0xF, false);
    return x + __int_as_float(yi);
}

// sum over each 16-lane row (both halves reduced independently)
__device__ __forceinline__ float rowsum16(float x) {
    x = rotadd16<1>(x);
    x = rotadd16<2>(x);
    x = rotadd16<4>(x);
    x = rotadd16<8>(x);
    return x;
}

// broadcast lane j's value within each 16-lane row (v_permlane16_b32)
__device__ __forceinline__ float bcast16(float x, int j) {
    int sel = (int)(0x11111111u * (unsigned)j);
    int yi = __builtin_amdgcn_permlane16(__float_as_int(x), __float_as_int(x),
                                         sel, sel, false, false);
    return __int_as_float(yi);
}

// v_rcp_f32 + 1 Newton step (~1-2 ulp; replaces full IEEE div sequence)
__device__ __forceinline__ float fastrcp(float x) {
    float r = __builtin_amdgcn_rcpf(x);
    return r * (2.0f - x * r);
}

// Fused: proj = z@W^T + b (+1e-6), then batched Householder QR (64x16 -> Q 64x16).
// One workgroup (8 waves) = 16 batch items, 64 KB LDS (5 WGs/WGP resident).
// geqr2 runs two items per wave in parallel (one per 16-lane half, 4 rows/lane,
// DPP-only reductions, LAPACK sign convention); then Q = A * R^{-1} via WMMA.
__launch_bounds__(256)
__global__ void QR_49263274885245_kernel(const float* __restrict__ z,
                                         const float* __restrict__ W,
                                         const float* __restrict__ bias,
                                         float* __restrict__ out)
{
    extern __shared__ float wlds[];      // 16 items * 64*16 floats = 64 KB
    const int tid   = threadIdx.x;
    const int wave  = tid >> 5;
    const int lane  = tid & 31;
    const int col16 = lane & 15;         // lane-in-half / N index / X column
    const int khalf = lane >> 4;         // wave half
    const int b0    = blockIdx.x * 16;   // first batch item of this group

    // ---------------- GEMM phase (WMMA f32 16x16x4, exact fp32) -------------
    const float* zrow = z + (size_t)(b0 + col16) * ZDIM + 2 * khalf;
    v2f bfrag[32];
    #pragma unroll
    for (int kk = 0; kk < 32; ++kk)
        bfrag[kk] = *(const v2f*)(zrow + 4 * kk);

    for (int t8 = 0; t8 < 8; ++t8) {
        const int t = wave * 8 + t8;     // tile t == QR row t
        const float* wrow = W + (size_t)(t * 16 + col16) * ZDIM + 2 * khalf;
        v8f acc = {0.f, 0.f, 0.f, 0.f, 0.f, 0.f, 0.f, 0.f};
        #pragma unroll
        for (int kk = 0; kk < 32; ++kk) {
            v2f af = *(const v2f*)(wrow + 4 * kk);
            acc = __builtin_amdgcn_wmma_f32_16x16x4_f32(
                      false, af, false, bfrag[kk], (short)0, acc, false, false);
        }
        const float* bp = bias + t * 16 + 8 * khalf;
        float* dst = &wlds[col16 * 1024 + t * 16 + 8 * khalf];
        #pragma unroll
        for (int v = 0; v < 8; ++v)
            dst[v] = acc[v] + bp[v] + 1e-6f;
    }
    __syncthreads();

    // ---------------- geqr2: two items per wave, one per 16-lane half -------
    v2f bf0[4], bf1[4];                  // B-fragments (R^{-1}) for items 0/1
    {
        const int item = wave * 2 + khalf;
        const float* mp = &wlds[item * 1024];
        float a0[16], a1[16], a2[16], a3[16];
        #pragma unroll
        for (int c4 = 0; c4 < 4; ++c4) {
            float4 r0 = *(const float4*)(mp + (col16     ) * 16 + 4 * c4);
            float4 r1 = *(const float4*)(mp + (col16 + 16) * 16 + 4 * c4);
            float4 r2 = *(const float4*)(mp + (col16 + 32) * 16 + 4 * c4);
            float4 r3 = *(const float4*)(mp + (col16 + 48) * 16 + 4 * c4);
            a0[4*c4+0]=r0.x; a0[4*c4+1]=r0.y; a0[4*c4+2]=r0.z; a0[4*c4+3]=r0.w;
            a1[4*c4+0]=r1.x; a1[4*c4+1]=r1.y; a1[4*c4+2]=r1.z; a1[4*c4+3]=r1.w;
            a2[4*c4+0]=r2.x; a2[4*c4+1]=r2.y; a2[4*c4+2]=r2.z; a2[4*c4+3]=r2.w;
            a3[4*c4+0]=r3.x; a3[4*c4+1]=r3.y; a3[4*c4+2]=r3.z; a3[4*c4+3]=r3.w;
        }

        #pragma unroll
        for (int j = 0; j < 16; ++j) {
            float x0 = (col16 >= j) ? a0[j] : 0.f;
            float ss = x0*x0 + a1[j]*a1[j] + a2[j]*a2[j] + a3[j]*a3[j];
            float nrm   = sqrtf(rowsum16(ss));
            float alpha = bcast16(a0[j], j);
            float beta  = (alpha >= 0.f) ? -nrm : nrm;   // LAPACK: -sign(alpha)*||x||
            float d     = alpha - beta;
            float tau   = -d * fastrcp(beta);            // (beta-alpha)/beta
            float scale = fastrcp(d);                    // 1/(alpha-beta)
            float v0 = (col16 > j) ? a0[j] * scale : ((col16 == j) ? 1.f : 0.f);
            float v1 = a1[j] * scale;
            float v2 = a2[j] * scale;
            float v3 = a3[j] * scale;
            if (col16 == j) a0[j] = beta;                // R[j][j]
            #pragma unroll
            for (int c = 0; c < 16; ++c) {
                if (c > j) {
                    float t = v0*a0[c] + v1*a1[c] + v2*a2[c] + v3*a3[c];
                    float tw = tau * rowsum16(t);
                    a0[c] -= tw * v0;
                    a1[c] -= tw * v1;
                    a2[c] -= tw * v2;
                    a3[c] -= tw * v3;
                }
            }
        }

        // X = R^{-1} backward substitution; lane computes column col16 of its item.
        float x[16];
        #pragma unroll
        for (int i = 15; i >= 0; --i) {
            float s = (col16 == i) ? 1.f : 0.f;
            #pragma unroll
            for (int k = 0; k < 16; ++k) {
                if (k > i) s -= bcast16(a0[k], i) * x[k];
            }
            x[i] = s * fastrcp(bcast16(a0[i], i));
        }

        // Build B-fragments for both items from registers: one cross-half
        // shuffle per value carries each half's column to the other half.
        #pragma unroll
        for (int kc = 0; kc < 4; ++kc) {
            #pragma unroll
            for (int j2 = 0; j2 < 2; ++j2) {
                float lo  = x[4*kc + j2];        // rows for khalf==0 readers
                float hi  = x[4*kc + 2 + j2];    // rows for khalf==1 readers
                float own = khalf ? hi : lo;     // X[4kc+2*khalf+j2] of own item
                float snd = khalf ? lo : hi;     // what the other half needs
                float sw  = __shfl_xor(snd, 16, 32);
                bf0[kc][j2] = khalf ? sw : own;  // item 0's fragment
                bf1[kc][j2] = khalf ? own : sw;  // item 1's fragment
            }
        }
    }

    // ---------------- Q = A * R^{-1} via WMMA, full wave per item -----------
    #pragma unroll
    for (int it = 0; it < 2; ++it) {
        const int item = wave * 2 + it;
        const float* mp = &wlds[item * 1024];
        const v2f* bf = it ? bf1 : bf0;
        const int b = b0 + item;
        #pragma unroll
        for (int rt = 0; rt < 4; ++rt) {
            v8f acc = {0.f, 0.f, 0.f, 0.f, 0.f, 0.f, 0.f, 0.f};
            #pragma unroll
            for (int kc = 0; kc < 4; ++kc) {
                v2f af = *(const v2f*)(mp + (16 * rt + col16) * 16 + 4 * kc + 2 * khalf);
                acc = __builtin_amdgcn_wmma_f32_16x16x4_f32(
                          false, af, false, bf[kc], (short)0, acc, false, false);
            }
            // lane holds Q[16*rt + v + 8*khalf][col16] for v = 0..7
            float* op = out + ((size_t)b * 64 + 16 * rt + 8 * khalf) * 16 + col16;
            #pragma unroll
            for (int v = 0; v < 8; ++v)
                op[v * 16] = acc[v];
        }
    }
}

extern "C" void kernel_launch(void* const* d_in, const int* in_sizes, int n_in,
                              void* d_out, int out_size, void* d_ws, size_t ws_size,
                              hipStream_t stream) {
    const float* z    = (const float*)d_in[0];
    const float* W    = (const float*)d_in[1];
    const float* bias = (const float*)d_in[2];
    float* out = (float*)d_out;
    const int B = in_sizes[0] / ZDIM;           // 65536
    dim3 grid(B / 16), block(256);
    hipLaunchKernelGGL(QR_49263274885245_kernel, grid, block,
                       16 * 64 * 16 * sizeof(float) /* 64 KB LDS */, stream,
                       z, W, bias, out);
}